// GNN_40690520163162
// MI455X (gfx1250) — compile-verified
//
#include <hip/hip_runtime.h>
#include <hip/hip_bf16.h>

typedef __attribute__((ext_vector_type(2))) float v2f;
typedef __attribute__((ext_vector_type(8))) float v8f;

#define N_NODES 50000
#define N_GRAPHS 2048

__device__ __forceinline__ void atomic_add_f32(float* p, float v) {
    unsafeAtomicAdd(p, v);   // hardware GLOBAL_ATOMIC_ADD_F32 (no CAS loop)
}

// ---------------- degree / normalization ----------------
__global__ void init_deg_kernel(float* deg, int n) {
    int i = blockIdx.x * blockDim.x + threadIdx.x;
    if (i < n) deg[i] = 1.0f;               // self loop contributes 1
}

__global__ void count_deg_kernel(const int* __restrict__ dst, float* deg, int E) {
    int i = blockIdx.x * blockDim.x + threadIdx.x;
    if (i < E) atomic_add_f32(&deg[dst[i]], 1.0f);
}

__global__ void dinv_kernel(float* deg, int n) {
    int i = blockIdx.x * blockDim.x + threadIdx.x;
    if (i < n) deg[i] = rsqrtf(deg[i]);     // deg >= 1 always
}

// ---------------- generic zero ----------------
__global__ void zero_kernel(float* p, int n) {
    int i = blockIdx.x * blockDim.x + threadIdx.x;
    if (i < n) p[i] = 0.0f;
}

// ---------------- fp32 WMMA GEMM: H[M,64] = A[M,K] @ W[K,64] (+bias,relu) ----
// blockDim = 256 (8 waves = 2 row-stripes x 4 col-tiles). W is staged once per
// block into LDS (B fragments -> ds_load_2addr). A rows are CLAMPED, not
// predicated: out-of-range rows only corrupt output rows that the store guard
// discards, so A loads stay unconditional b64 (no exec-mask churn). K is a
// template parameter so the V_WMMA_F32_16X16X4_F32 chain fully unrolls.
template <int K>
__global__ void gemm_wmma_f32(const float* __restrict__ A,
                              const float* __restrict__ W,
                              const float* __restrict__ bias,  // may be null
                              float* __restrict__ H,
                              int M, int do_relu) {
    __shared__ float Wlds[K * 64];
    #pragma unroll 4
    for (int i = threadIdx.x; i < K * 64; i += 256)
        Wlds[i] = W[i];
    __syncthreads();

    const int wave = threadIdx.x >> 5;        // 0..7
    const int lane = threadIdx.x & 31;
    const int half = lane >> 4;               // 0: lanes 0-15, 1: lanes 16-31
    const int l16  = lane & 15;
    const int col0 = (wave & 3) * 16;         // column tile
    const int row0 = blockIdx.x * 32 + (wave >> 2) * 16;
    const int m    = row0 + l16;
    const int mc   = (m < M) ? m : (M - 1);   // clamp: safe, store-guarded
    const float* __restrict__ Arow = A + (size_t)mc * K + 2 * half;
    const float* __restrict__ Brow = Wlds + 2 * half * 64 + col0 + l16;

    v8f acc = {};
    #pragma unroll
    for (int k0 = 0; k0 < K; k0 += 4) {
        // A 16x4 tile: lane-half holds K pair {k0+2h, k0+2h+1} (ISA 7.12.2)
        v2f a = *(const v2f*)(Arow + k0);
        // B 4x16 tile from LDS: row K striped across lanes, halves hold pairs
        v2f b;
        b.x = Brow[k0 * 64];
        b.y = Brow[k0 * 64 + 64];
        acc = __builtin_amdgcn_wmma_f32_16x16x4_f32(
            /*neg_a=*/false, a, /*neg_b=*/false, b,
            /*c_mod=*/(short)0, acc, /*reuse_a=*/false, /*reuse_b=*/false);
    }
    const float bv = bias ? bias[col0 + l16] : 0.0f;
    #pragma unroll
    for (int r = 0; r < 8; ++r) {
        int mm = row0 + r + 8 * half;         // D layout: VGPR r -> rows r / r+8
        if (mm < M) {
            float v = acc[r] + bv;
            if (do_relu) v = fmaxf(v, 0.0f);
            H[(size_t)mm * 64 + col0 + l16] = v;
        }
    }
}

// ---------------- edge scatter: agg[dst] += dinv[s]*dinv[d]*h[s] -------------
// one wave32 per edge; wave-uniform indices forced into SGPRs so the only
// vector traffic is the 256B coalesced feature gather + the f32 atomics.
__global__ void edge_agg_kernel(const int* __restrict__ src,
                                const int* __restrict__ dst,
                                const float* __restrict__ dinv,
                                const float* __restrict__ h,
                                float* __restrict__ agg, int E) {
    int e = __builtin_amdgcn_readfirstlane(blockIdx.x * (blockDim.x >> 5) +
                                           (threadIdx.x >> 5));
    if (e >= E) return;
    int lane = threadIdx.x & 31;
    int s = __builtin_amdgcn_readfirstlane(src[e]);
    int d = __builtin_amdgcn_readfirstlane(dst[e]);
    float nrm = dinv[s] * dinv[d];
    const float2* hs = (const float2*)(h + (size_t)s * 64);
    float2 v = hs[lane];
    float* ad = agg + (size_t)d * 64 + 2 * lane;
    atomic_add_f32(ad + 0, nrm * v.x);
    atomic_add_f32(ad + 1, nrm * v.y);
}

// ---------------- epilogue: self-loop + bias + relu --------------------------
__global__ void finalize_kernel(const float* __restrict__ agg,
                                const float* __restrict__ hlin,
                                const float* __restrict__ dinv,
                                const float* __restrict__ bias,
                                float* __restrict__ hout, int total) {
    int i = blockIdx.x * blockDim.x + threadIdx.x;
    if (i >= total) return;
    int n = i >> 6, f = i & 63;
    float di = dinv[n];
    float v = agg[i] + di * di * hlin[i] + bias[f];
    hout[i] = fmaxf(v, 0.0f);
}

// ---------------- mean pooling ----------------
__global__ void pool_acc_kernel(const float* __restrict__ h,
                                const int* __restrict__ batch,
                                float* __restrict__ pooled,
                                float* __restrict__ counts, int n) {
    int node = __builtin_amdgcn_readfirstlane(blockIdx.x * (blockDim.x >> 5) +
                                              (threadIdx.x >> 5));
    if (node >= n) return;
    int lane = threadIdx.x & 31;
    int g = __builtin_amdgcn_readfirstlane(batch[node]);
    const float2* hn = (const float2*)(h + (size_t)node * 64);
    float2 v = hn[lane];
    float* pg = pooled + (size_t)g * 64 + 2 * lane;
    atomic_add_f32(pg + 0, v.x);
    atomic_add_f32(pg + 1, v.y);
    if (lane == 0) atomic_add_f32(&counts[g], 1.0f);
}

__global__ void pool_div_kernel(float* pooled, const float* counts, int total) {
    int i = blockIdx.x * blockDim.x + threadIdx.x;
    if (i >= total) return;
    float c = fmaxf(counts[i >> 6], 1.0f);
    pooled[i] = pooled[i] / c;
}

// ---------------- final 64->2 head ----------------
__global__ void head_kernel(const float* __restrict__ r1,
                            const float* __restrict__ lw2,
                            const float* __restrict__ lb2,
                            float* __restrict__ out, int G) {
    int i = blockIdx.x * blockDim.x + threadIdx.x;
    if (i >= G * 2) return;
    int g = i >> 1, j = i & 1;
    float acc = lb2[j];
    #pragma unroll
    for (int k = 0; k < 64; ++k) acc += r1[(size_t)g * 64 + k] * lw2[k * 2 + j];
    out[i] = acc;
}

static inline size_t alignup(size_t v) { return (v + 255) & ~(size_t)255; }

extern "C" void kernel_launch(void* const* d_in, const int* in_sizes, int n_in,
                              void* d_out, int out_size, void* d_ws, size_t ws_size,
                              hipStream_t stream) {
    const float* x     = (const float*)d_in[0];
    const int*   eidx  = (const int*)d_in[1];
    const int*   batch = (const int*)d_in[2];
    const float* W1 = (const float*)d_in[3];  const float* b1 = (const float*)d_in[4];
    const float* W2 = (const float*)d_in[5];  const float* b2 = (const float*)d_in[6];
    const float* W3 = (const float*)d_in[7];  const float* b3 = (const float*)d_in[8];
    const float* lw1 = (const float*)d_in[9]; const float* lb1 = (const float*)d_in[10];
    const float* lw2 = (const float*)d_in[11]; const float* lb2 = (const float*)d_in[12];
    float* out = (float*)d_out;

    const int N = N_NODES, G = N_GRAPHS;
    const int E = in_sizes[1] / 2;
    const int* src = eidx;       // edge_index[0]
    const int* dst = eidx + E;   // edge_index[1]

    // workspace carve-up
    char* w = (char*)d_ws;
    float* dinv   = (float*)w; w += alignup((size_t)N * 4);
    float* hlin   = (float*)w; w += alignup((size_t)N * 64 * 4);
    float* agg    = (float*)w; w += alignup((size_t)N * 64 * 4);
    float* hfeat  = (float*)w; w += alignup((size_t)N * 64 * 4);
    float* pooled = (float*)w; w += alignup((size_t)G * 64 * 4);
    float* counts = (float*)w; w += alignup((size_t)G * 4);
    float* relu1  = (float*)w; w += alignup((size_t)G * 64 * 4);
    (void)ws_size; (void)n_in;

    const int NF = N * 64;
    dim3 blk256(256);
    int gN    = (N + 255) / 256;
    int gNF   = (NF + 255) / 256;
    int gE    = (E + 255) / 256;
    int gE8   = (E + 7) / 8;       // 8 waves/block, one edge per wave
    int gN8   = (N + 7) / 8;
    int gGemmN = (N + 31) / 32;    // 32 rows per block
    int gGemmG = (G + 31) / 32;

    // degrees -> dinv
    init_deg_kernel<<<gN, blk256, 0, stream>>>(dinv, N);
    count_deg_kernel<<<gE, blk256, 0, stream>>>(dst, dinv, E);
    dinv_kernel<<<gN, blk256, 0, stream>>>(dinv, N);

    // ---- layer 1 (K=32) ----
    gemm_wmma_f32<32><<<gGemmN, blk256, 0, stream>>>(x, W1, nullptr, hlin, N, 0);
    zero_kernel<<<gNF, blk256, 0, stream>>>(agg, NF);
    edge_agg_kernel<<<gE8, blk256, 0, stream>>>(src, dst, dinv, hlin, agg, E);
    finalize_kernel<<<gNF, blk256, 0, stream>>>(agg, hlin, dinv, b1, hfeat, NF);

    // ---- layer 2 (K=64) ----
    gemm_wmma_f32<64><<<gGemmN, blk256, 0, stream>>>(hfeat, W2, nullptr, hlin, N, 0);
    zero_kernel<<<gNF, blk256, 0, stream>>>(agg, NF);
    edge_agg_kernel<<<gE8, blk256, 0, stream>>>(src, dst, dinv, hlin, agg, E);
    finalize_kernel<<<gNF, blk256, 0, stream>>>(agg, hlin, dinv, b2, hfeat, NF);

    // ---- layer 3 (K=64) ----
    gemm_wmma_f32<64><<<gGemmN, blk256, 0, stream>>>(hfeat, W3, nullptr, hlin, N, 0);
    zero_kernel<<<gNF, blk256, 0, stream>>>(agg, NF);
    edge_agg_kernel<<<gE8, blk256, 0, stream>>>(src, dst, dinv, hlin, agg, E);
    finalize_kernel<<<gNF, blk256, 0, stream>>>(agg, hlin, dinv, b3, hfeat, NF);

    // ---- mean pool ----
    int GF = G * 64;
    zero_kernel<<<(GF + 255) / 256, blk256, 0, stream>>>(pooled, GF);
    zero_kernel<<<(G + 255) / 256, blk256, 0, stream>>>(counts, G);
    pool_acc_kernel<<<gN8, blk256, 0, stream>>>(hfeat, batch, pooled, counts, N);
    pool_div_kernel<<<(GF + 255) / 256, blk256, 0, stream>>>(pooled, counts, GF);

    // ---- MLP head ----
    gemm_wmma_f32<64><<<gGemmG, blk256, 0, stream>>>(pooled, lw1, lb1, relu1, G, 1);
    head_kernel<<<(G * 2 + 255) / 256, blk256, 0, stream>>>(relu1, lw2, lb2, out, G);
    (void)out_size;
}